// FP16ToFP32Converter_12137577578806
// MI455X (gfx1250) — compile-verified
//
#include <hip/hip_runtime.h>
#include <stdint.h>

// FP16 bit-pattern (as 0.0/1.0 floats) -> FP32 bit-pattern (as 0.0/1.0 floats).
// Memory-bound streaming kernel: 64B in + 128B out per element, ~805MB total
// => ~34.5us floor at 23.3 TB/s. Compute is trivial integer bit logic.
// CDNA5-specific path: per-lane GLOBAL_LOAD_ASYNC_TO_LDS_B128 staging of the
// input (tracked with ASYNCcnt / s_wait_asynccnt), non-temporal b128 stores.

typedef float v4f __attribute__((ext_vector_type(4)));
typedef int   v4i_vs __attribute__((vector_size(16)));  // matches builtin param

#if defined(__has_builtin)
#if __has_builtin(__builtin_amdgcn_global_load_async_to_lds_b128) && \
    __has_builtin(__builtin_amdgcn_s_wait_asynccnt)
#define FP16CVT_USE_ASYNC_LDS 1
#endif
#endif
#ifndef FP16CVT_USE_ASYNC_LDS
#define FP16CVT_USE_ASYNC_LDS 0
#endif

__device__ __forceinline__ unsigned bit_of(float f) {
  // f is exactly 0.0f or 1.0f; 1.0f == 0x3F800000 -> bit 23 is the value.
  return (__builtin_bit_cast(unsigned, f) >> 23) & 1u;
}

__global__ __launch_bounds__(256) void fp16_bits_to_fp32_bits_kernel(
    const float* __restrict__ in, float* __restrict__ out, int n) {
  const int tid = threadIdx.x;
  const long long idx = (long long)blockIdx.x * 256 + tid;
  if (idx >= (long long)n) return;

  const float* gp = in + idx * 16;

  v4f a0, a1, a2, a3;
#if FP16CVT_USE_ASYNC_LDS
  // Stage this wave's input tile through LDS with the gfx1250 async copy path.
  // Each lane copies its own 64B (4 x b128); ASYNCcnt tracks completion, and
  // each lane only reads LDS it filled itself, so s_wait_asynccnt(0) suffices
  // (no workgroup barrier required).
  __shared__ v4f tile[256 * 4];  // 16 KB per 256-thread block
  v4f* lp = &tile[tid * 4];
#pragma unroll
  for (int j = 0; j < 4; ++j) {
    __builtin_amdgcn_global_load_async_to_lds_b128(
        (v4i_vs*)(gp + j * 4),
        (v4i_vs*)(lp + j),
        /*imm offset=*/0, /*cpol=*/0);
  }
  __builtin_amdgcn_s_wait_asynccnt(0);
  a0 = lp[0];
  a1 = lp[1];
  a2 = lp[2];
  a3 = lp[3];
#else
  const v4f* gv = (const v4f*)gp;
  a0 = __builtin_nontemporal_load(gv + 0);
  a1 = __builtin_nontemporal_load(gv + 1);
  a2 = __builtin_nontemporal_load(gv + 2);
  a3 = __builtin_nontemporal_load(gv + 3);
#endif

  // Fields (MSB-first, matching the reference's bit order):
  //   a0 = { s, e4, e3, e2 }, a1 = { e1, e0, m[0], m[1] },
  //   a2 = { m[2..5] },       a3 = { m[6..9] }
  const unsigned sb = bit_of(a0.x);
  const unsigned e  = (bit_of(a0.y) << 4) | (bit_of(a0.z) << 3) |
                      (bit_of(a0.w) << 2) | (bit_of(a1.x) << 1) |
                      bit_of(a1.y);

  // e==0  -> zero / subnormal: exponent and mantissa fully zeroed.
  // e==31 -> inf/NaN: exponent 255, mantissa passes through (0 for inf).
  // else  -> exponent e+112 (fp32 bias 127 - fp16 bias 15), mantissa through.
  const unsigned keep = (e != 0u) ? 1u : 0u;
  const unsigned expb = (e == 0u) ? 0u : ((e == 31u) ? 255u : (e + 112u));

  v4f o0 = { (float)sb,
             (float)((expb >> 7) & 1u),
             (float)((expb >> 6) & 1u),
             (float)((expb >> 5) & 1u) };
  v4f o1 = { (float)((expb >> 4) & 1u),
             (float)((expb >> 3) & 1u),
             (float)((expb >> 2) & 1u),
             (float)((expb >> 1) & 1u) };
  v4f o2 = { (float)(expb & 1u),
             (float)(keep & bit_of(a1.z)),
             (float)(keep & bit_of(a1.w)),
             (float)(keep & bit_of(a2.x)) };
  v4f o3 = { (float)(keep & bit_of(a2.y)),
             (float)(keep & bit_of(a2.z)),
             (float)(keep & bit_of(a2.w)),
             (float)(keep & bit_of(a3.x)) };
  v4f o4 = { (float)(keep & bit_of(a3.y)),
             (float)(keep & bit_of(a3.z)),
             (float)(keep & bit_of(a3.w)),
             0.0f };
  v4f zz = { 0.0f, 0.0f, 0.0f, 0.0f };

  // 8 x 16B = 128B per element, written once and never re-read -> NT stores
  // (output stream is 512MB, far beyond the 192MB L2).
  v4f* ov = (v4f*)(out + idx * 32);
  __builtin_nontemporal_store(o0, ov + 0);
  __builtin_nontemporal_store(o1, ov + 1);
  __builtin_nontemporal_store(o2, ov + 2);
  __builtin_nontemporal_store(o3, ov + 3);
  __builtin_nontemporal_store(o4, ov + 4);
  __builtin_nontemporal_store(zz, ov + 5);
  __builtin_nontemporal_store(zz, ov + 6);
  __builtin_nontemporal_store(zz, ov + 7);
}

extern "C" void kernel_launch(void* const* d_in, const int* in_sizes, int n_in,
                              void* d_out, int out_size, void* d_ws, size_t ws_size,
                              hipStream_t stream) {
  (void)n_in; (void)out_size; (void)d_ws; (void)ws_size;
  const float* in = (const float*)d_in[0];
  float* out = (float*)d_out;
  const int n = in_sizes[0] / 16;          // 4,194,304 elements
  const int blocks = (n + 255) / 256;      // 16,384 blocks of 8 wave32s
  fp16_bits_to_fp32_bits_kernel<<<blocks, 256, 0, stream>>>(in, out, n);
}